// Atk_Poisson_71674414236196
// MI455X (gfx1250) — compile-verified
//
#include <hip/hip_runtime.h>

// Integrate-and-fire scan over T: mem += x[t]; spike = (mem > 1); mem -= spike.
// Pure HBM-streaming workload: ~327 MB of traffic, ~14 us floor at 23.3 TB/s.
// fp32 kept exactly (threshold compare on accumulated sums is precision-critical).
// CDNA5 paths: async global->LDS b128 loads (ASYNCcnt, in-order completion, so
// incremental s_wait_asynccnt pipelines compute under outstanding loads),
// ds_load_b128 tile reads, non-temporal b128 stores (stream > 192MB L2).

typedef float vfloat4 __attribute__((ext_vector_type(4)));
typedef int   v4i     __attribute__((ext_vector_type(4)));

typedef __attribute__((address_space(1))) v4i* gv4i_ptr;   // global (AS1)
typedef __attribute__((address_space(3))) v4i* lv4i_ptr;   // LDS    (AS3)

#define THRESH 1.0f
#define T_FIXED 8
#define BLOCK 256

#if defined(__has_builtin)
#  if __has_builtin(__builtin_amdgcn_global_load_async_to_lds_b128)
#    define HAVE_ASYNC_LDS 1
#  else
#    define HAVE_ASYNC_LDS 0
#  endif
#  if __has_builtin(__builtin_amdgcn_s_wait_asynccnt)
#    define HAVE_WAIT_ASYNC 1
#  else
#    define HAVE_WAIT_ASYNC 0
#  endif
#else
#  define HAVE_ASYNC_LDS 0
#  define HAVE_WAIT_ASYNC 0
#endif

template <int CNT>
__device__ __forceinline__ void wait_asynccnt() {
#if HAVE_WAIT_ASYNC
  __builtin_amdgcn_s_wait_asynccnt(CNT);
#else
  asm volatile("s_wait_asynccnt %0" ::"i"(CNT) : "memory");
#endif
}

__global__ __launch_bounds__(BLOCK) void if_scan8_vec4(
    const float* __restrict__ x, const float* __restrict__ mem0,
    float* __restrict__ out, int n4) {
  const int i = blockIdx.x * BLOCK + (int)threadIdx.x;
  const bool active = (i < n4);
  const vfloat4* __restrict__ xv  = (const vfloat4*)x;
  const vfloat4* __restrict__ m0v = (const vfloat4*)mem0;
  vfloat4* __restrict__ ov        = (vfloat4*)out;

#if HAVE_ASYNC_LDS
  // 8 timesteps * 256 lanes * 16B = 32 KB LDS per block.
  __shared__ vfloat4 tile[T_FIXED][BLOCK];
  if (active) {
#pragma unroll
    for (int t = 0; t < T_FIXED; ++t) {
      __builtin_amdgcn_global_load_async_to_lds_b128(
          (gv4i_ptr)(xv + (size_t)t * n4 + i),
          (lv4i_ptr)&tile[t][threadIdx.x],
          /*offset=*/0, /*cpol=*/0);
    }
  }
#endif

  if (!active) return;

  vfloat4 mem = __builtin_nontemporal_load(m0v + i);

  // One scan step; async loads complete in ASYNCcnt order, so waiting for
  // count <= T-1-t guarantees timestep t's tile is resident in LDS while
  // later timesteps are still streaming in.
#if HAVE_ASYNC_LDS
#define IF_STEP(t)                                                      \
  do {                                                                  \
    wait_asynccnt<T_FIXED - 1 - (t)>();                                 \
    vfloat4 xt = tile[(t)][threadIdx.x]; /* ds_load_b128 */             \
    vfloat4 s;                                                          \
    mem += xt;                                                          \
    s.x = mem.x > THRESH ? 1.0f : 0.0f;                                 \
    s.y = mem.y > THRESH ? 1.0f : 0.0f;                                 \
    s.z = mem.z > THRESH ? 1.0f : 0.0f;                                 \
    s.w = mem.w > THRESH ? 1.0f : 0.0f;                                 \
    mem -= s;                                                           \
    __builtin_nontemporal_store(s, ov + (size_t)(t) * n4 + i);          \
  } while (0)
#else
#define IF_STEP(t)                                                      \
  do {                                                                  \
    vfloat4 xt = __builtin_nontemporal_load(xv + (size_t)(t) * n4 + i); \
    vfloat4 s;                                                          \
    mem += xt;                                                          \
    s.x = mem.x > THRESH ? 1.0f : 0.0f;                                 \
    s.y = mem.y > THRESH ? 1.0f : 0.0f;                                 \
    s.z = mem.z > THRESH ? 1.0f : 0.0f;                                 \
    s.w = mem.w > THRESH ? 1.0f : 0.0f;                                 \
    mem -= s;                                                           \
    __builtin_nontemporal_store(s, ov + (size_t)(t) * n4 + i);          \
  } while (0)
#endif

  IF_STEP(0); IF_STEP(1); IF_STEP(2); IF_STEP(3);
  IF_STEP(4); IF_STEP(5); IF_STEP(6); IF_STEP(7);
#undef IF_STEP
}

// Generic fallback: any T, scalar elements in [start, n).
__global__ __launch_bounds__(BLOCK) void if_scan_generic(
    const float* __restrict__ x, const float* __restrict__ mem0,
    float* __restrict__ out, long long start, long long n, int T) {
  long long i = start + blockIdx.x * (long long)BLOCK + threadIdx.x;
  if (i >= n) return;
  float mem = mem0[i];
  for (int t = 0; t < T; ++t) {
    mem += x[(long long)t * n + i];
    float s = mem > THRESH ? 1.0f : 0.0f;
    mem -= s;
    out[(long long)t * n + i] = s;
  }
}

extern "C" void kernel_launch(void* const* d_in, const int* in_sizes, int n_in,
                              void* d_out, int out_size, void* d_ws, size_t ws_size,
                              hipStream_t stream) {
  (void)n_in; (void)out_size; (void)d_ws; (void)ws_size;
  const float* x    = (const float*)d_in[0];
  const float* mem0 = (const float*)d_in[1];
  float* out        = (float*)d_out;

  const long long N = (long long)in_sizes[1];           // B*C*H*W
  const int T = (int)((long long)in_sizes[0] / N);      // time steps

  if (T == T_FIXED) {
    const long long n4ll = N / 4;
    const int n4 = (int)n4ll;
    const int blocks = (n4 + BLOCK - 1) / BLOCK;
    if (blocks > 0)
      if_scan8_vec4<<<dim3(blocks), dim3(BLOCK), 0, stream>>>(x, mem0, out, n4);
    const long long tail = N - n4ll * 4;                // N % 4 leftover scalars
    if (tail > 0) {
      const int tblocks = (int)((tail + BLOCK - 1) / BLOCK);
      if_scan_generic<<<dim3(tblocks), dim3(BLOCK), 0, stream>>>(
          x, mem0, out, n4ll * 4, N, T);
    }
  } else {
    const int blocks = (int)((N + BLOCK - 1) / BLOCK);
    if_scan_generic<<<dim3(blocks), dim3(BLOCK), 0, stream>>>(x, mem0, out, 0, N, T);
  }
}